// SimpleHGNConv_45475113730251
// MI455X (gfx1250) — compile-verified
//
#include <hip/hip_runtime.h>
#include <hip/hip_bf16.h>

// ---------------- problem constants (match reference) ----------------
#define NN   100000
#define EE   800000
#define INF_ 256
#define DD   64
#define HH   4
#define EFF  64
#define NETT 8
#define SLOPE 0.2f

typedef __attribute__((ext_vector_type(2))) float v2f;
typedef __attribute__((ext_vector_type(8))) float v8f;

// ---------------- ordered-uint encoding for float atomic max ----------------
__device__ __forceinline__ unsigned ordf(float f) {
    unsigned u = __float_as_uint(f);
    return (u & 0x80000000u) ? ~u : (u | 0x80000000u);
}
__device__ __forceinline__ float unordf(unsigned o) {
    unsigned u = (o & 0x80000000u) ? (o & 0x7FFFFFFFu) : ~o;
    return __uint_as_float(u);
}

__device__ __forceinline__ float nan_guard(float v) {
    // jnp.nan_to_num(nan=0.0): NaN->0, +/-inf -> +/-FLT_MAX
    return (v != v) ? 0.0f
                    : fminf(fmaxf(v, -3.4028234663852886e38f),
                            3.4028234663852886e38f);
}

// reduce across each 16-lane half of the wave (masks < 16 never cross halves)
__device__ __forceinline__ float half_reduce16(float v) {
    v += __shfl_xor(v, 1);
    v += __shfl_xor(v, 2);
    v += __shfl_xor(v, 4);
    v += __shfl_xor(v, 8);
    return v;
}

// ======================================================================
// Kernel 1: fused dual GEMM with V_WMMA_F32_16X16X4_F32, 4-tile register
// blocking (one wave = 16 nodes x 64 cols = one full head).
//   blockIdx.y == 0: emb = nan_to_num(h @ W) -> ws, PLUS fused per-head
//                    attention pre-dots h_l/h_r (completed in-wave).
//   blockIdx.y == 1: d_out = h @ res_W + res_b
// Block = 128 threads (4 waves); the 4 waves cover all 256 output cols.
// ======================================================================
__global__ void __launch_bounds__(128)
hgn_gemm_kernel(const float* __restrict__ h, const float* __restrict__ W,
                const float* __restrict__ resW, const float* __restrict__ resb,
                const float* __restrict__ a_l, const float* __restrict__ a_r,
                float* __restrict__ emb, float* __restrict__ h_l,
                float* __restrict__ h_r, float* __restrict__ outres) {
    const int wave  = threadIdx.x >> 5;            // 0..3  == head index
    const int lane  = threadIdx.x & 31;
    const bool isRes = blockIdx.y != 0;
    const int c0    = wave << 6;                   // column base (64 per wave)
    const int node0 = blockIdx.x << 4;             // N divisible by 16
    const int mrow  = lane & 15;
    const int khalf = lane >> 4;                   // 0 or 1

    const float* __restrict__ B    = isRes ? resW : W;
    const float* __restrict__ Arow = h + (size_t)(node0 + mrow) * INF_;
    const float* __restrict__ Bc   = B + (c0 + mrow);

    v8f acc0 = {}, acc1 = {}, acc2 = {}, acc3 = {};
#pragma unroll 2
    for (int k = 0; k < INF_; k += 4) {
        const int ka = k + (khalf << 1);
        v2f a;  a.x  = Arow[ka];                       a.y  = Arow[ka + 1];
        v2f b0; b0.x = Bc[(size_t)ka * 256 +  0];      b0.y = Bc[(size_t)(ka+1) * 256 +  0];
        v2f b1; b1.x = Bc[(size_t)ka * 256 + 16];      b1.y = Bc[(size_t)(ka+1) * 256 + 16];
        v2f b2; b2.x = Bc[(size_t)ka * 256 + 32];      b2.y = Bc[(size_t)(ka+1) * 256 + 32];
        v2f b3; b3.x = Bc[(size_t)ka * 256 + 48];      b3.y = Bc[(size_t)(ka+1) * 256 + 48];
        acc0 = __builtin_amdgcn_wmma_f32_16x16x4_f32(false, a, false, b0, (short)0, acc0, false, false);
        acc1 = __builtin_amdgcn_wmma_f32_16x16x4_f32(false, a, false, b1, (short)0, acc1, false, false);
        acc2 = __builtin_amdgcn_wmma_f32_16x16x4_f32(false, a, false, b2, (short)0, acc2, false, false);
        acc3 = __builtin_amdgcn_wmma_f32_16x16x4_f32(false, a, false, b3, (short)0, acc3, false, false);
    }

    if (isRes) {
#pragma unroll
        for (int j = 0; j < 4; ++j) {
            const v8f* accp = (j == 0) ? &acc0 : (j == 1) ? &acc1 : (j == 2) ? &acc2 : &acc3;
            const int n = c0 + (j << 4) + mrow;
            const float bias = resb[n];
#pragma unroll
            for (int r = 0; r < 8; ++r) {
                const int m = r + (khalf << 3);
                outres[(size_t)(node0 + m) * 256 + n] = (*accp)[r] + bias;
            }
        }
    } else {
        float pl[8] = {0,0,0,0,0,0,0,0};
        float pr[8] = {0,0,0,0,0,0,0,0};
#pragma unroll
        for (int j = 0; j < 4; ++j) {
            const v8f* accp = (j == 0) ? &acc0 : (j == 1) ? &acc1 : (j == 2) ? &acc2 : &acc3;
            const int n = c0 + (j << 4) + mrow;     // flat a_l/a_r index == n
            const float alv = a_l[n];
            const float arv = a_r[n];
#pragma unroll
            for (int r = 0; r < 8; ++r) {
                const int m = r + (khalf << 3);
                const float v = nan_guard((*accp)[r]);
                emb[(size_t)(node0 + m) * 256 + n] = v;
                pl[r] += alv * v;
                pr[r] += arv * v;
            }
        }
        // complete the per-head dot: sum over the 16 lanes of each half-wave
#pragma unroll
        for (int r = 0; r < 8; ++r) {
            pl[r] = half_reduce16(pl[r]);
            pr[r] = half_reduce16(pr[r]);
        }
        if (mrow == 0) {                            // lane 0 (rows 0-7), lane 16 (rows 8-15)
#pragma unroll
            for (int r = 0; r < 8; ++r) {
                const int m = r + (khalf << 3);
                h_l[(size_t)(node0 + m) * HH + wave] = pl[r];
                h_r[(size_t)(node0 + m) * HH + wave] = pr[r];
            }
        }
    }
}

// ======================================================================
// Kernel 2: edge-type attention term  h_e [NET,H]  (tiny, 32 threads)
// ======================================================================
__global__ void hgn_he_kernel(const float* __restrict__ edge_emb,
                              const float* __restrict__ W_e,
                              const float* __restrict__ a_e,
                              float* __restrict__ h_e) {
    const int t = threadIdx.x;
    if (t >= NETT * HH) return;
    const int net = t >> 2, hh = t & 3;
    float s = 0.f;
    for (int f = 0; f < EFF; ++f) {
        float ev = 0.f;
        for (int j = 0; j < EFF; ++j)
            ev += edge_emb[net * EFF + j] * W_e[(size_t)j * (EFF*HH) + hh * EFF + f];
        s += a_e[hh * EFF + f] * ev;
    }
    h_e[t] = s;
}

// ======================================================================
// Kernel 3: edge logits + leaky-relu + segment max (ordered-uint atomicMax)
// ======================================================================
__global__ void __launch_bounds__(256)
hgn_logits_kernel(const int* __restrict__ row, const int* __restrict__ col,
                  const int* __restrict__ tp,
                  const float* __restrict__ h_l, const float* __restrict__ h_r,
                  const float* __restrict__ h_e,
                  float* __restrict__ attbuf, unsigned* __restrict__ m_ord) {
    const int t = blockIdx.x * blockDim.x + threadIdx.x;
    if (t >= EE * HH) return;
    const int e = t >> 2, hh = t & 3;
    const int r = row[e], c = col[e], ty = tp[e];
    float x = h_l[r * HH + hh] + h_r[c * HH + hh] + h_e[ty * HH + hh];
    x = x > 0.f ? x : SLOPE * x;
    attbuf[t] = x;
    atomicMax(&m_ord[c * HH + hh], ordf(x));
}

// ======================================================================
// Kernel 4: ex = exp(att - m[col]); segment sum of ex into denom
// ======================================================================
__global__ void __launch_bounds__(256)
hgn_exp_kernel(const int* __restrict__ col, const unsigned* __restrict__ m_ord,
               float* __restrict__ attbuf, float* __restrict__ denom) {
    const int t = blockIdx.x * blockDim.x + threadIdx.x;
    if (t >= EE * HH) return;
    const int e = t >> 2, hh = t & 3;
    const int c = col[e];
    const float m  = unordf(m_ord[c * HH + hh]);
    const float ex = __expf(attbuf[t] - m);
    attbuf[t] = ex;                               // reuse buffer for ex
    unsafeAtomicAdd(&denom[c * HH + hh], ex);     // global_atomic_add_f32
}

// ======================================================================
// Kernel 5: gather-multiply-scatter. 64 threads per edge, float4 gather,
// 4 hardware f32 atomic adds into d_out (which already holds the residual).
// ======================================================================
__global__ void __launch_bounds__(256)
hgn_scatter_kernel(const int* __restrict__ row, const int* __restrict__ col,
                   const float* __restrict__ emb, const float* __restrict__ attbuf,
                   const float* __restrict__ denom, float* __restrict__ out) {
    const long long t = (long long)blockIdx.x * blockDim.x + threadIdx.x;
    if (t >= (long long)EE * 64) return;
    const int e  = (int)(t >> 6);
    const int q  = (int)(t & 63);                 // 4-float chunk id within H*D=256
    const int hh = q >> 4;                        // 16 chunks per head
    const int r = row[e], c = col[e];
    const float alpha = attbuf[e * HH + hh] / denom[c * HH + hh];
    const float4 v = *(const float4*)(emb + (size_t)r * 256 + q * 4);
    float* base = out + (size_t)c * 256 + q * 4;
    unsafeAtomicAdd(base + 0, v.x * alpha);
    unsafeAtomicAdd(base + 1, v.y * alpha);
    unsafeAtomicAdd(base + 2, v.z * alpha);
    unsafeAtomicAdd(base + 3, v.w * alpha);
}

// ======================================================================
// Kernel 6: in-place ELU on d_out, float4-vectorized (b128 load/store)
// ======================================================================
__global__ void __launch_bounds__(256)
hgn_elu_kernel(float4* __restrict__ out) {
    const size_t t = (size_t)blockIdx.x * blockDim.x + threadIdx.x;
    if (t >= (size_t)NN * 64) return;             // N*256/4 float4 elements
    float4 v = out[t];
    v.x = v.x > 0.f ? v.x : (__expf(v.x) - 1.0f);
    v.y = v.y > 0.f ? v.y : (__expf(v.y) - 1.0f);
    v.z = v.z > 0.f ? v.z : (__expf(v.z) - 1.0f);
    v.w = v.w > 0.f ? v.w : (__expf(v.w) - 1.0f);
    out[t] = v;
}

// ======================================================================
extern "C" void kernel_launch(void* const* d_in, const int* in_sizes, int n_in,
                              void* d_out, int out_size, void* d_ws, size_t ws_size,
                              hipStream_t stream) {
    const float* h        = (const float*)d_in[0];
    const float* W        = (const float*)d_in[1];
    const float* W_e      = (const float*)d_in[2];
    const float* edge_emb = (const float*)d_in[3];
    const float* a_l      = (const float*)d_in[4];
    const float* a_r      = (const float*)d_in[5];
    const float* a_e      = (const float*)d_in[6];
    const float* res_W    = (const float*)d_in[7];
    const float* res_b    = (const float*)d_in[8];
    const int*   row      = (const int*)d_in[9];
    const int*   col      = (const int*)d_in[10];
    const int*   tp       = (const int*)d_in[11];
    float*       out      = (float*)d_out;

    // ---- workspace carve-up (256B aligned) ----
    char* ws = (char*)d_ws;
    auto align256 = [](size_t x) { return (x + 255) & ~(size_t)255; };
    size_t off = 0;
    float*    emb    = (float*)(ws + off); off += align256((size_t)NN * DD * HH * 4);
    float*    h_l    = (float*)(ws + off); off += align256((size_t)NN * HH * 4);
    float*    h_r    = (float*)(ws + off); off += align256((size_t)NN * HH * 4);
    float*    h_e    = (float*)(ws + off); off += align256((size_t)NETT * HH * 4);
    unsigned* m_ord  = (unsigned*)(ws + off); off += align256((size_t)NN * HH * 4);
    float*    denom  = (float*)(ws + off); off += align256((size_t)NN * HH * 4);
    float*    attbuf = (float*)(ws + off); off += align256((size_t)EE * HH * 4);

    // segment-reduction state: 0 is the minimum ordered-uint; denom starts at 0
    hipMemsetAsync(m_ord, 0, (size_t)NN * HH * 4, stream);
    hipMemsetAsync(denom, 0, (size_t)NN * HH * 4, stream);

    // 1) dual GEMM + fused h_l/h_r: emb -> ws ; d_out = h@res_W + res_b
    hgn_gemm_kernel<<<dim3(NN / 16, 2), 128, 0, stream>>>(
        h, W, res_W, res_b, a_l, a_r, emb, h_l, h_r, out);

    // 2) per-edge-type attention term (tiny)
    hgn_he_kernel<<<1, 32, 0, stream>>>(edge_emb, W_e, a_e, h_e);

    // 3) logits + segment max
    hgn_logits_kernel<<<(EE * HH + 255) / 256, 256, 0, stream>>>(
        row, col, tp, h_l, h_r, h_e, attbuf, m_ord);

    // 4) exp + segment sum
    hgn_exp_kernel<<<(EE * HH + 255) / 256, 256, 0, stream>>>(col, m_ord, attbuf, denom);

    // 5) message scatter into d_out (on top of residual)
    long long sc_threads = (long long)EE * 64;
    hgn_scatter_kernel<<<(unsigned)((sc_threads + 255) / 256), 256, 0, stream>>>(
        row, col, emb, attbuf, denom, out);

    // 6) ELU in place (float4)
    hgn_elu_kernel<<<(unsigned)(((size_t)NN * 64 + 255) / 256), 256, 0, stream>>>(
        (float4*)out);
}